// ST_BIFNeuron_MS_30794915512543
// MI455X (gfx1250) — compile-verified
//
#include <hip/hip_runtime.h>

// ST-BIF spiking neuron, multi-step forward.
//   input  [T*B, TOKENS, C] fp32 -> viewed as [T=8, B=16, 197, 768]
//   out[t] = spike(t) * q_threshold, spike in {-1,0,1}, sequential scan over T.
// Memory-bound: 155 MB traffic -> ~6.7 us floor @ 23.3 TB/s on MI455X.

#define T_STEPS  8
#define NB       16
#define TOKENS   197
#define CCH      768
#define D_FLAT   (NB * TOKENS * CCH)   /* 2,420,736 floats per time step */
#define D4       (D_FLAT / 4)          /* 605,184 float4 groups          */
#define BLOCK    256                   /* 8 wave32 per workgroup         */
#define POS_MAX  15.0f
#define EFFECT_T 7

// gfx1250 async global->LDS path. The probe confirmed the builtin exists and
// takes (int4 AS1*, int4 AS3*, imm offset, imm cpol).
#if defined(__has_builtin)
#  if __has_builtin(__builtin_amdgcn_global_load_async_to_lds_b128)
#    define HAVE_ASYNC_LDS 1
#  endif
#endif
#ifndef HAVE_ASYNC_LDS
#  define HAVE_ASYNC_LDS 0
#endif

typedef int v4i __attribute__((vector_size(16)));
typedef __attribute__((address_space(1))) v4i* gv4i_p;   // global int4*
typedef __attribute__((address_space(3))) v4i* lv4i_p;   // LDS int4*

__global__ __launch_bounds__(BLOCK) void st_bif_ms_kernel(
    const float* __restrict__ in,     // [T*B, TOKENS, C] flat
    const float* __restrict__ bias,   // [C]
    const float* __restrict__ ba,     // [6] biasAllocator
    const float* __restrict__ qp,     // [1] q_threshold
    float* __restrict__ out)          // same shape as in
{
  // Keep mul+add and the divide un-fused / exact: spike decisions are hard
  // thresholds and 1-ulp drift flips discrete spikes.
  #pragma clang fp contract(off)

  __shared__ float s_bias[CCH];
  const int tid = threadIdx.x;

  // ---- stage bias_channel (3 KB) into LDS --------------------------------
#if HAVE_ASYNC_LDS
  if (tid < CCH / 4) {
    __builtin_amdgcn_global_load_async_to_lds_b128(
        (gv4i_p)(const_cast<float*>(bias) + tid * 4),
        (lv4i_p)(&s_bias[tid * 4]),
        /*imm offset*/ 0, /*cpol*/ 0);
  }
  asm volatile("s_wait_asynccnt 0" ::: "memory");
  __syncthreads();
#else
  if (tid < CCH / 4) {
    reinterpret_cast<float4*>(s_bias)[tid] =
        reinterpret_cast<const float4*>(bias)[tid];
  }
  __syncthreads();
#endif

  // ---- alloc weights (uniform -> scalar regs) ----------------------------
  // alloc[0] = 1 - sum(biasAllocator) (left-to-right sum), alloc[1..6] = ba.
  float asum = ba[0];
  asum = asum + ba[1];
  asum = asum + ba[2];
  asum = asum + ba[3];
  asum = asum + ba[4];
  asum = asum + ba[5];
  float alloc[EFFECT_T];
  alloc[0] = 1.0f - asum;
  alloc[1] = ba[0]; alloc[2] = ba[1]; alloc[3] = ba[2];
  alloc[4] = ba[3]; alloc[5] = ba[4]; alloc[6] = ba[5];
  const float q = qp[0];

  // ---- per-thread element group ------------------------------------------
  const int gid  = blockIdx.x * BLOCK + tid;       // < D4 exactly (no tail)
  const int base = gid * 4;                        // flat float index in [0,D)
  const int c    = base % CCH;                     // channel index (16B aligned)

  const float4 b4 = *reinterpret_cast<const float4*>(&s_bias[c]);

  // gfx1250 prefetch hint for the t=0 stream (emits global_prefetch_b8).
  __builtin_prefetch(in + base, 0, 3);

  // Issue all 8 time-step loads up front: 8-deep memory-level parallelism.
  float4 xa[T_STEPS];
  #pragma unroll
  for (int t = 0; t < T_STEPS; ++t) {
    xa[t] = *reinterpret_cast<const float4*>(in + (size_t)t * D_FLAT + base);
  }

  float v0 = 0.f, v1 = 0.f, v2 = 0.f, v3 = 0.f;     // membrane (prefire = 0)
  float a0 = 0.f, a1 = 0.f, a2 = 0.f, a3 = 0.f;     // spike tracer

  #pragma unroll
  for (int t = 0; t < T_STEPS; ++t) {
    float x0 = xa[t].x, x1 = xa[t].y, x2 = xa[t].z, x3 = xa[t].w;
    if (t < EFFECT_T) {
      const float al = alloc[t];
      x0 = x0 + al * b4.x;
      x1 = x1 + al * b4.y;
      x2 = x2 + al * b4.z;
      x3 = x3 + al * b4.w;
    }
    // exact IEEE divide to match the reference's x / q_threshold
    x0 = x0 / q;  x1 = x1 / q;  x2 = x2 / q;  x3 = x3 / q;

    v0 += x0;  v1 += x1;  v2 += x2;  v3 += x3;

    const float s0 = (v0 >= 1.0f && a0 < POS_MAX) ? 1.0f
                   : ((v0 < 0.0f && a0 > 0.0f) ? -1.0f : 0.0f);
    const float s1 = (v1 >= 1.0f && a1 < POS_MAX) ? 1.0f
                   : ((v1 < 0.0f && a1 > 0.0f) ? -1.0f : 0.0f);
    const float s2 = (v2 >= 1.0f && a2 < POS_MAX) ? 1.0f
                   : ((v2 < 0.0f && a2 > 0.0f) ? -1.0f : 0.0f);
    const float s3 = (v3 >= 1.0f && a3 < POS_MAX) ? 1.0f
                   : ((v3 < 0.0f && a3 > 0.0f) ? -1.0f : 0.0f);

    v0 -= s0;  v1 -= s1;  v2 -= s2;  v3 -= s3;      // soft reset, v_th = 1
    a0 += s0;  a1 += s1;  a2 += s2;  a3 += s3;      // tracer update

    float4 o;
    o.x = s0 * q;  o.y = s1 * q;  o.z = s2 * q;  o.w = s3 * q;
    *reinterpret_cast<float4*>(out + (size_t)t * D_FLAT + base) = o;
  }
}

extern "C" void kernel_launch(void* const* d_in, const int* in_sizes, int n_in,
                              void* d_out, int out_size, void* d_ws, size_t ws_size,
                              hipStream_t stream) {
  (void)in_sizes; (void)n_in; (void)out_size; (void)d_ws; (void)ws_size;
  const float* in   = (const float*)d_in[0];   // [T*B, TOKENS, C] fp32
  const float* bias = (const float*)d_in[1];   // [C]
  const float* ba   = (const float*)d_in[2];   // [6]
  const float* q    = (const float*)d_in[3];   // scalar
  float* out        = (float*)d_out;

  static_assert(D4 % BLOCK == 0, "exact grid, no tail");
  st_bif_ms_kernel<<<dim3(D4 / BLOCK), dim3(BLOCK), 0, stream>>>(
      in, bias, ba, q, out);
}